// SPConv2d_16621523436018
// MI455X (gfx1250) — compile-verified
//
#include <hip/hip_runtime.h>
#include <hip/hip_bf16.h>

typedef __attribute__((ext_vector_type(2))) float v2f;
typedef __attribute__((ext_vector_type(8))) float v8f;

namespace {
constexpr int Bn = 8;    // batch
constexpr int Cn = 64;   // input channels (GEMM K)
constexpr int On = 64;   // output channels (GEMM M)
constexpr int Hn = 128;
constexpr int Wn = 128;
constexpr int WT = 32;        // tile width
constexpr int HT = 2;         // tile height (rows per block) -> halo amp 2x
constexpr int NP = WT * HT;   // 64 positions per block (GEMM N)
constexpr int XSW = WT + 2;   // 34
constexpr int XSH = HT + 2;   // 4
constexpr int ST  = 68;       // selT row stride: conflict-free b64 B reads
}

__global__ __launch_bounds__(256) void spconv_fused_wmma(
    const float* __restrict__ x,
    const float* __restrict__ core,
    const float* __restrict__ periphery,
    const float* __restrict__ thr,
    const float* __restrict__ scl,
    float* __restrict__ out)
{
    __shared__ float xs[Cn * XSH * XSW];   // 34816 B halo tile
    __shared__ float selT[NP * ST];        // 17408 B, [position][channel]
    __shared__ float divs[NP];
    __shared__ float msk[NP];

    const int tid = threadIdx.x;
    const int w0  = blockIdx.x * WT;
    const int h0  = blockIdx.y * HT;
    const int b   = blockIdx.z;

    const int wave = tid >> 5;
    const int lane = tid & 31;
    const int lrow = lane & 15;
    const bool hi  = lane >= 16;

    // ---- A fragments live in registers: same for every block, L2-resident ----
    const int mt   = wave >> 1;        // M tile 0..3
    const int nt0  = (wave & 1) * 2;   // first of two N tiles
    const int arow = mt * 16 + lrow;
    const int ka0  = hi ? 2 : 0;       // ISA 16x4 f32 A-layout K split
    v2f afrag[16];
    #pragma unroll
    for (int ks = 0; ks < 16; ++ks)
        afrag[ks] = *(const v2f*)&core[arow * Cn + ka0 + ks * 4];

    if (tid < NP) divs[tid] = 0.0f;

    // ---- halo tile load (zero-pad outside image) ----
    for (int i = tid; i < Cn * XSH * XSW; i += 256) {
        int c  = i / (XSH * XSW);
        int r  = (i / XSW) % XSH;
        int cc = i % XSW;
        int gh = h0 - 1 + r;
        int gw = w0 - 1 + cc;
        float v = 0.0f;
        if ((unsigned)gh < (unsigned)Hn && (unsigned)gw < (unsigned)Wn)
            v = x[(((b * Cn + c) * Hn) + gh) * Wn + gw];
        xs[i] = v;
    }
    __syncthreads();

    // periphery order = taps k=0..8 skipping center (k=4)
    float per[8];
    #pragma unroll
    for (int k = 0; k < 8; ++k) per[k] = periphery[k];

    // ---- phase 2: agg + partial div per (c, position); reduce div over c ----
    const int p  = tid & 63;   // position in tile: p = rr*32 + wi
    const int cg = tid >> 6;   // channel group 0..3
    const int wi = p & 31;
    const int rr = p >> 5;
    float divPart = 0.0f;
    #pragma unroll
    for (int j = 0; j < 16; ++j) {
        const int c = cg * 16 + j;
        const float* base = &xs[(c * XSH + rr) * XSW + wi];
        const float ctr = base[XSW + 1];
        float agg = 0.0f;
        int pk = 0;
        #pragma unroll
        for (int di = 0; di < 3; ++di) {
            #pragma unroll
            for (int dj = 0; dj < 3; ++dj) {
                if (di == 1 && dj == 1) continue;
                const float t = base[di * XSW + dj];
                const float d = t - ctr;
                divPart += d * d;
                agg += t * per[pk];
                ++pk;
            }
        }
        selT[p * ST + c] = agg;   // stash agg in transposed sel buffer
    }
    atomicAdd(&divs[p], divPart);   // ds_add_f32, 4 adds per slot
    __syncthreads();

    // ---- mask per position: sigmoid((div - thr)*scale) > 0.5 ----
    if (tid < NP) {
        const float z = (divs[tid] - thr[0]) * scl[0];
        const float s = 1.0f / (1.0f + expf(-z));
        msk[tid] = (s > 0.5f) ? 1.0f : 0.0f;
    }
    __syncthreads();

    // ---- sel = mask ? agg : center, in place (transposed layout) ----
    #pragma unroll
    for (int j = 0; j < 16; ++j) {
        const int c = cg * 16 + j;
        const float ctr = xs[(c * XSH + rr + 1) * XSW + wi + 1];
        const float a = selT[p * ST + c];
        selT[p * ST + c] = (msk[p] != 0.0f) ? a : ctr;
    }
    __syncthreads();

    // ---- phase 3: D(16x16) += A(16x4) x B(4x16), K=64, two N tiles/wave ----
    v8f acc0 = {};
    v8f acc1 = {};
    const float* b0base = &selT[(nt0 * 16 + lrow) * ST + ka0];
    const float* b1base = &selT[((nt0 + 1) * 16 + lrow) * ST + ka0];
    #pragma unroll
    for (int ks = 0; ks < 16; ++ks) {
        const v2f b0 = *(const v2f*)&b0base[ks * 4];   // ds_load_b64, imm offset
        const v2f b1 = *(const v2f*)&b1base[ks * 4];
        acc0 = __builtin_amdgcn_wmma_f32_16x16x4_f32(
            false, afrag[ks], false, b0, (short)0, acc0, false, false);
        acc1 = __builtin_amdgcn_wmma_f32_16x16x4_f32(
            false, afrag[ks], false, b1, (short)0, acc1, false, false);
    }

    // ---- store both D tiles: VGPR r -> rows (r, r+8) of the 16x16 tile ----
    #pragma unroll
    for (int t = 0; t < 2; ++t) {
        const int nt = nt0 + t;
        const int p2 = nt * 16 + lrow;
        const int hh = h0 + (p2 >> 5);
        const int ww = w0 + (p2 & 31);
        #pragma unroll
        for (int r = 0; r < 8; ++r) {
            const int m = mt * 16 + r + (hi ? 8 : 0);
            out[(((b * On + m) * Hn) + hh) * Wn + ww] = (t == 0) ? acc0[r] : acc1[r];
        }
    }
}

extern "C" void kernel_launch(void* const* d_in, const int* in_sizes, int n_in,
                              void* d_out, int out_size, void* d_ws, size_t ws_size,
                              hipStream_t stream) {
    const float* x         = (const float*)d_in[0];
    const float* core      = (const float*)d_in[1];
    const float* periphery = (const float*)d_in[2];
    const float* thr       = (const float*)d_in[3];
    const float* scl       = (const float*)d_in[4];
    // d_in[5]=stride, d_in[6]=padding: fixed at 1 for this problem shape
    float* out = (float*)d_out;

    dim3 grid(Wn / WT, Hn / HT, Bn);   // 4 x 64 x 8 = 2048 workgroups
    spconv_fused_wmma<<<grid, 256, 0, stream>>>(x, core, periphery, thr, scl, out);
}